// GCNModel_88759794139304
// MI455X (gfx1250) — compile-verified
//
#include <hip/hip_runtime.h>
#include <hip/hip_bf16.h>

// ---------------------------------------------------------------------------
// GCN 4-layer forward for MI455X (gfx1250), wave32.
// Edge scatter/gather is the memory-bound part (fits largely in 192MB L2);
// the per-node linears run on the matrix pipe via V_WMMA_F32_16X16X4_F32
// with compile-time layer dims: straight-line WMMAs, branchless clamped
// operand loads, and a uniform full-tile fast path for stores.
// ---------------------------------------------------------------------------

typedef __attribute__((ext_vector_type(2))) float v2f;
typedef __attribute__((ext_vector_type(8))) float v8f;

#define LEN_SCALE_INV (1.0f / 20000.0f)

// --- degree init: deg (stored in dinv buffer) = 1.0 (self loop) -------------
__global__ void k_deg_init(float* __restrict__ dinv, int N) {
    int i = blockIdx.x * blockDim.x + threadIdx.x;
    if (i < N) dinv[i] = 1.0f;
}

// --- degree accumulate over dst ---------------------------------------------
__global__ void k_deg_edges(const int* __restrict__ dst, float* __restrict__ dinv,
                            int E) {
    int e = blockIdx.x * blockDim.x + threadIdx.x;
    if (e < E) unsafeAtomicAdd(&dinv[dst[e]], 1.0f);
}

// --- dinv_sqrt = rsqrt(deg); dinv = 1/deg; x0 = read_length/20000 -----------
__global__ void k_deg_fin(const int* __restrict__ read_length,
                          float* __restrict__ dinv, float* __restrict__ dsq,
                          float* __restrict__ x0, int N) {
    int i = blockIdx.x * blockDim.x + threadIdx.x;
    if (i < N) {
        float deg = dinv[i];
        dsq[i]  = rsqrtf(deg);
        dinv[i] = 1.0f / deg;
        x0[i]   = (float)read_length[i] * LEN_SCALE_INV;
    }
}

// --- norm_e = dinv_sqrt[src] * dinv_sqrt[dst] -------------------------------
__global__ void k_norm(const int* __restrict__ src, const int* __restrict__ dst,
                       const float* __restrict__ dsq, float* __restrict__ nrm,
                       int E) {
    int e = blockIdx.x * blockDim.x + threadIdx.x;
    if (e < E) nrm[e] = dsq[src[e]] * dsq[dst[e]];
}

// --- zero fill ---------------------------------------------------------------
__global__ void k_zero(float* __restrict__ p, long long n) {
    long long i = (long long)blockIdx.x * blockDim.x + threadIdx.x;
    if (i < n) p[i] = 0.0f;
}

// --- xw = x @ W via V_WMMA_F32_16X16X4_F32 -----------------------------------
// One wave per 16(M) x 16(N) output tile. K unrolled at compile time.
// A layout (ISA 7.12.2, 32-bit A 16x4): lanes 0-15 hold K={0,1} in the two
// VGPRs, lanes 16-31 hold K={2,3}. B (4x16): VGPR j, lane-half g -> K=g*2+j,
// N = lane&15. C/D: VGPR v, lanes 0-15 -> M=v, lanes 16-31 -> M=v+8.
// Operand loads are branchless (clamped address + v_cndmask select) so EXEC
// stays all-1s through the WMMA chain; stores take a wave-uniform full-tile
// fast path (8 unguarded stores off one base with immediate offsets).
template <int F_IN, int F_OUT>
__global__ void k_linear_wmma(const float* __restrict__ x,
                              const float* __restrict__ W,
                              float* __restrict__ xw, int N) {
    constexpr int TILES_N = (F_OUT + 15) / 16;
    constexpr int KCH     = (F_IN + 3) / 4;

    const int wave   = blockIdx.x * (blockDim.x >> 5) + (threadIdx.x >> 5);
    const int lane   = threadIdx.x & 31;
    const int tile_m = wave / TILES_N;
    const int tile_n = wave - tile_m * TILES_N;
    const int row0   = tile_m * 16;
    if (row0 >= N) return;  // wave-uniform

    const int  lg   = lane >> 4;   // lane half 0/1
    const int  lmod = lane & 15;
    const int  m    = row0 + lmod;
    const int  mc   = (m < N) ? m : (N - 1);          // clamped A row
    const int  nc   = tile_n * 16 + lmod;             // output column
    const bool nok  = (nc < F_OUT);
    const int  ncc  = nok ? nc : 0;                   // clamped B/D column

    v8f acc = {0.f, 0.f, 0.f, 0.f, 0.f, 0.f, 0.f, 0.f};

#pragma unroll
    for (int kc = 0; kc < KCH; ++kc) {
        const int k0  = kc * 4 + lg * 2;
        const int k1  = k0 + 1;
        const int k0c = (k0 < F_IN) ? k0 : (F_IN - 1);
        const int k1c = (k1 < F_IN) ? k1 : (F_IN - 1);

        // A: branchless clamped loads + select
        float a0 = x[(long long)mc * F_IN + k0c];
        float a1 = x[(long long)mc * F_IN + k1c];
        a0 = (m < N && k0 < F_IN) ? a0 : 0.f;
        a1 = (m < N && k1 < F_IN) ? a1 : 0.f;
        v2f a; a.x = a0; a.y = a1;

        // B: branchless clamped loads + select
        float b0 = W[k0c * F_OUT + ncc];
        float b1 = W[k1c * F_OUT + ncc];
        b0 = (nok && k0 < F_IN) ? b0 : 0.f;
        b1 = (nok && k1 < F_IN) ? b1 : 0.f;
        v2f b; b.x = b0; b.y = b1;

        // (neg_a, A, neg_b, B, c_mod, C, reuse_a, reuse_b)
        acc = __builtin_amdgcn_wmma_f32_16x16x4_f32(
            false, a, false, b, (short)0, acc, false, false);
    }

    const bool fullM = (row0 + 16 <= N);  // wave-uniform (always true: N%16==0)
    if (nok) {
        float* p = xw + (long long)(row0 + lg * 8) * F_OUT + nc;
        if (fullM) {
            // fast path: 8 unguarded stores, immediate offsets off one base
#pragma unroll
            for (int v = 0; v < 8; ++v) p[(long long)v * F_OUT] = acc[v];
        } else {
#pragma unroll
            for (int v = 0; v < 8; ++v) {
                const int mrow = row0 + v + lg * 8;
                if (mrow < N) p[(long long)v * F_OUT] = acc[v];
            }
        }
    }
}

// --- edge scatter: one thread per (edge, feature) ----------------------------
template <int LOGF>
__global__ void k_scatter(const float* __restrict__ xw,
                          const int* __restrict__ src,
                          const int* __restrict__ dst,
                          const float* __restrict__ nrm,
                          float* __restrict__ agg, long long EF) {
    const long long idx = (long long)blockIdx.x * blockDim.x + threadIdx.x;
    if (idx >= EF) return;
    const int e = (int)(idx >> LOGF);
    const int f = (int)(idx & ((1 << LOGF) - 1));
    const int s = src[e];
    const int d = dst[e];
    const float v = xw[((long long)s << LOGF) + f] * nrm[e];
    unsafeAtomicAdd(&agg[((long long)d << LOGF) + f], v);
}

// --- out = agg + xw * dinv + b ----------------------------------------------
template <int LOGF>
__global__ void k_finalize(const float* __restrict__ agg,
                           const float* __restrict__ xw,
                           const float* __restrict__ dinv,
                           const float* __restrict__ bias,
                           float* __restrict__ out, long long NF) {
    const long long i = (long long)blockIdx.x * blockDim.x + threadIdx.x;
    if (i >= NF) return;
    const int node = (int)(i >> LOGF);
    const int f    = (int)(i & ((1 << LOGF) - 1));
    out[i] = agg[i] + xw[i] * dinv[node] + bias[f];
}

// --- per-layer driver --------------------------------------------------------
template <int F_IN, int F_OUT, int LOGF>
static void run_layer(const float* xin, const float* W, const float* bias,
                      float* xw, float* agg, const float* dinv,
                      const int* src, const int* dst, const float* nrm,
                      float* xout, int N, int E, hipStream_t stream) {
    constexpr int T = 256;
    constexpr int TILES_N = (F_OUT + 15) / 16;

    const long long waves  = (long long)((N + 15) / 16) * TILES_N;
    const long long blocks = (waves + 7) / 8;  // 8 waves (256 thr) per block
    k_linear_wmma<F_IN, F_OUT><<<(unsigned)blocks, 256, 0, stream>>>(xin, W, xw, N);

    const long long nf = (long long)N * F_OUT;
    k_zero<<<(unsigned)((nf + T - 1) / T), T, 0, stream>>>(agg, nf);

    const long long ef = (long long)E * F_OUT;
    k_scatter<LOGF><<<(unsigned)((ef + T - 1) / T), T, 0, stream>>>(
        xw, src, dst, nrm, agg, ef);

    k_finalize<LOGF><<<(unsigned)((nf + T - 1) / T), T, 0, stream>>>(
        agg, xw, dinv, bias, xout, nf);
}

extern "C" void kernel_launch(void* const* d_in, const int* in_sizes, int n_in,
                              void* d_out, int out_size, void* d_ws, size_t ws_size,
                              hipStream_t stream) {
    const int* read_length = (const int*)d_in[0];
    const int* edge        = (const int*)d_in[1];
    const int  N = in_sizes[0];
    const int  E = in_sizes[1] / 2;
    const int* src = edge;
    const int* dst = edge + E;

    const float* W1 = (const float*)d_in[2]; const float* B1 = (const float*)d_in[3];
    const float* W2 = (const float*)d_in[4]; const float* B2 = (const float*)d_in[5];
    const float* W3 = (const float*)d_in[6]; const float* B3 = (const float*)d_in[7];
    const float* W4 = (const float*)d_in[8]; const float* B4 = (const float*)d_in[9];

    // workspace layout (floats): x[N*32] | xw[N*32] | agg[N*32] | dinv[N] |
    //                            dsq[N] | norm[E]   (~228 MB total)
    float* ws   = (float*)d_ws;
    float* xbuf = ws;
    float* xw   = xbuf + (size_t)N * 32;
    float* agg  = xw   + (size_t)N * 32;
    float* dinv = agg  + (size_t)N * 32;
    float* dsq  = dinv + N;
    float* nrm  = dsq  + N;

    const int T = 256;

    // degree (with self loop) -> dinv_sqrt, dinv, and x0
    k_deg_init<<<(N + T - 1) / T, T, 0, stream>>>(dinv, N);
    k_deg_edges<<<(E + T - 1) / T, T, 0, stream>>>(dst, dinv, E);
    k_deg_fin<<<(N + T - 1) / T, T, 0, stream>>>(read_length, dinv, dsq, xbuf, N);
    k_norm<<<(E + T - 1) / T, T, 0, stream>>>(src, dst, dsq, nrm, E);

    // 4 GCN layers; x ping-pongs in xbuf, last layer writes d_out
    run_layer<1, 4, 2>(xbuf, W1, B1, xw, agg, dinv, src, dst, nrm, xbuf, N, E, stream);
    run_layer<4, 8, 3>(xbuf, W2, B2, xw, agg, dinv, src, dst, nrm, xbuf, N, E, stream);
    run_layer<8, 16, 4>(xbuf, W3, B3, xw, agg, dinv, src, dst, nrm, xbuf, N, E, stream);
    run_layer<16, 32, 5>(xbuf, W4, B4, xw, agg, dinv, src, dst, nrm,
                         (float*)d_out, N, E, stream);
}